// GCN_86509231276868
// MI455X (gfx1250) — compile-verified
//
#include <hip/hip_runtime.h>
#include <hip/hip_bf16.h>

typedef float v2f __attribute__((ext_vector_type(2)));
typedef float v8f __attribute__((ext_vector_type(8)));

#define N0_   200000
#define N1_   50000
#define N2_   10000
#define E1_   750000
#define E2_   100000
#define IN_   128
#define HID_  256
#define OUT_  47
#define OUTP_ 48          // OUT_ padded to a multiple of 16

// ---- workspace layout (float offsets) ----
// [deg_src1 | deg_dst1 | deg_src2 | deg_dst2 | agg1 | agg2 | h | W2p | b2p]
#define OFF_DEG_SRC1  0
#define OFF_DEG_DST1  (OFF_DEG_SRC1 + N0_)
#define OFF_DEG_SRC2  (OFF_DEG_DST1 + N1_)
#define OFF_DEG_DST2  (OFF_DEG_SRC2 + N1_)
#define OFF_AGG1      (OFF_DEG_DST2 + N2_)
#define OFF_AGG2      (OFF_AGG1 + (size_t)N1_ * IN_)
#define OFF_H         (OFF_AGG2 + (size_t)N2_ * HID_)
#define OFF_W2P       (OFF_H + (size_t)N1_ * HID_)
#define OFF_B2P       (OFF_W2P + (size_t)HID_ * OUTP_)
#define ZERO_COUNT    OFF_H   // everything before h must start at 0

__global__ void gcn_zero_f32(float* p, int n) {
    int i = blockIdx.x * blockDim.x + threadIdx.x;
    if (i < n) p[i] = 0.0f;
}

__global__ void gcn_degree(const int* __restrict__ idx, float* __restrict__ deg, int n) {
    int i = blockIdx.x * blockDim.x + threadIdx.x;
    if (i < n) atomicAdd(&deg[idx[i]], 1.0f);
}

// Pad W2 (HID_ x OUT_) -> W2p (HID_ x OUTP_), b2 (OUT_) -> b2p (OUTP_), zero-filled.
__global__ void gcn_pad_w2(const float* __restrict__ W2, const float* __restrict__ b2,
                           float* __restrict__ W2p, float* __restrict__ b2p) {
    int i = blockIdx.x * blockDim.x + threadIdx.x;
    if (i < HID_ * OUTP_) {
        int r = i / OUTP_, c = i - r * OUTP_;
        W2p[i] = (c < OUT_) ? W2[r * OUT_ + c] : 0.0f;
    } else if (i < HID_ * OUTP_ + OUTP_) {
        int c = i - HID_ * OUTP_;
        b2p[c] = (c < OUT_) ? b2[c] : 0.0f;
    }
}

// agg[dst] += feat[src] * rsqrt(max(deg_src,1)); COLS/4 lanes per edge, 4 floats each
template<int COLS>
__global__ void gcn_scatter(const float* __restrict__ feat,
                            const int* __restrict__ src,
                            const int* __restrict__ dst,
                            const float* __restrict__ deg_src,
                            float* __restrict__ agg, int n_edges) {
    const int lanes_per_edge = COLS / 4;
    int tid = blockIdx.x * blockDim.x + threadIdx.x;
    int e = tid / lanes_per_edge;
    if (e >= n_edges) return;
    int c4 = (tid % lanes_per_edge) * 4;
    int s = src[e];
    int d = dst[e];
    float ns = rsqrtf(fmaxf(deg_src[s], 1.0f));
    const float4 v = *(const float4*)(feat + (size_t)s * COLS + c4);
    float* ap = agg + (size_t)d * COLS + c4;
    atomicAdd(ap + 0, v.x * ns);
    atomicAdd(ap + 1, v.y * ns);
    atomicAdd(ap + 2, v.z * ns);
    atomicAdd(ap + 3, v.w * ns);
}

// mat[r, :] *= rsqrt(max(deg[r],1)); cols = 1<<shift
__global__ void gcn_scale_rows(float* __restrict__ mat, const float* __restrict__ deg,
                               int n_elems, int shift) {
    int i = blockIdx.x * blockDim.x + threadIdx.x;
    if (i >= n_elems) return;
    int r = i >> shift;
    mat[i] *= rsqrtf(fmaxf(deg[r], 1.0f));
}

// C[M,Nc] = act(A[M,K] @ B[K,NB] + bias[NB]); one 16x16 tile per wave32.
// K and NB are compile-time -> fully unrolled K-loop, immediate-offset loads,
// no divergence in the hot loop (B is padded so every bcol < NB is in-bounds).
// v_wmma_f32_16x16x4_f32: A lane l -> row (l&15), K-pair (l>>4)*2;
// B/C/D lane l -> col (l&15), upper half-lanes rows +2 (B) / +8 (C/D).
template<int K, int NB, bool RELU>
__global__ void __launch_bounds__(256)
gcn_gemm_wmma(const float* __restrict__ A,
              const float* __restrict__ B,
              const float* __restrict__ bias,
              float* __restrict__ C,
              int Nc, int tilesN, int totalTiles) {
    int wave = threadIdx.x >> 5;
    int lane = threadIdx.x & 31;
    int tile = blockIdx.x * (blockDim.x >> 5) + wave;
    if (tile >= totalTiles) return;           // wave-uniform: EXEC stays all-1s

    int tm = tile / tilesN;
    int tn = tile - tm * tilesN;
    int l15  = lane & 15;
    int half = lane >> 4;

    int bcol = tn * 16 + l15;                 // always < NB (B padded)
    const float* Ap = A + (size_t)(tm * 16 + l15) * K + half * 2;
    const float* Bp = B + (size_t)(half * 2) * NB + bcol;

    v8f c = {};
    #pragma unroll
    for (int k0 = 0; k0 < K; k0 += 4) {
        v2f a = *(const v2f*)(Ap + k0);       // 8B-aligned
        v2f b;
        b.x = Bp[k0 * NB];
        b.y = Bp[k0 * NB + NB];
        c = __builtin_amdgcn_wmma_f32_16x16x4_f32(
                false, a, false, b, (short)0, c, false, false);
    }

    if (bcol < Nc) {                          // only GEMM2 cols 47 drop out
        float bv = bias[bcol];
        #pragma unroll
        for (int r = 0; r < 8; ++r) {
            int m = tm * 16 + r + half * 8;
            float v = c[r] + bv;
            if (RELU) v = fmaxf(v, 0.0f);
            C[(size_t)m * Nc + bcol] = v;
        }
    }
}

extern "C" void kernel_launch(void* const* d_in, const int* in_sizes, int n_in,
                              void* d_out, int out_size, void* d_ws, size_t ws_size,
                              hipStream_t stream) {
    const float* x    = (const float*)d_in[0];
    const float* W1   = (const float*)d_in[1];
    const float* b1   = (const float*)d_in[2];
    const float* W2   = (const float*)d_in[3];
    const float* b2   = (const float*)d_in[4];
    const int*   src1 = (const int*)d_in[5];
    const int*   dst1 = (const int*)d_in[6];
    const int*   src2 = (const int*)d_in[7];
    const int*   dst2 = (const int*)d_in[8];

    float* ws       = (float*)d_ws;
    float* deg_src1 = ws + OFF_DEG_SRC1;
    float* deg_dst1 = ws + OFF_DEG_DST1;
    float* deg_src2 = ws + OFF_DEG_SRC2;
    float* deg_dst2 = ws + OFF_DEG_DST2;
    float* agg1     = ws + OFF_AGG1;
    float* agg2     = ws + OFF_AGG2;
    float* h        = ws + OFF_H;
    float* W2p      = ws + OFF_W2P;
    float* b2p      = ws + OFF_B2P;
    float* out      = (float*)d_out;

    const int T = 256;

    // 1) zero degree + aggregation buffers
    {
        int n = (int)ZERO_COUNT;
        gcn_zero_f32<<<(n + T - 1) / T, T, 0, stream>>>(ws, n);
    }

    // 1b) pad W2/b2 to 48 columns
    {
        int n = HID_ * OUTP_ + OUTP_;
        gcn_pad_w2<<<(n + T - 1) / T, T, 0, stream>>>(W2, b2, W2p, b2p);
    }

    // 2) degrees
    gcn_degree<<<(E1_ + T - 1) / T, T, 0, stream>>>(src1, deg_src1, E1_);
    gcn_degree<<<(E1_ + T - 1) / T, T, 0, stream>>>(dst1, deg_dst1, E1_);
    gcn_degree<<<(E2_ + T - 1) / T, T, 0, stream>>>(src2, deg_src2, E2_);
    gcn_degree<<<(E2_ + T - 1) / T, T, 0, stream>>>(dst2, deg_dst2, E2_);

    // 3) layer-1 edge scatter: agg1[dst1] += x[src1] * norm_src
    {
        long long work = (long long)E1_ * (IN_ / 4);
        gcn_scatter<IN_><<<(int)((work + T - 1) / T), T, 0, stream>>>(
            x, src1, dst1, deg_src1, agg1, E1_);
    }

    // 4) agg1 *= norm_dst
    {
        int n = N1_ * IN_;
        gcn_scale_rows<<<(n + T - 1) / T, T, 0, stream>>>(agg1, deg_dst1, n, 7);
    }

    // 5) h = relu(agg1 @ W1 + b1)   [M=50000, K=128, NB=Nc=256]
    {
        int tilesM = N1_ / 16, tilesN = HID_ / 16;
        int total = tilesM * tilesN;
        int blocks = (total + (T / 32) - 1) / (T / 32);
        gcn_gemm_wmma<IN_, HID_, true><<<blocks, T, 0, stream>>>(
            agg1, W1, b1, h, HID_, tilesN, total);
    }

    // 6) layer-2 edge scatter: agg2[dst2] += h[src2] * norm_src
    {
        long long work = (long long)E2_ * (HID_ / 4);
        gcn_scatter<HID_><<<(int)((work + T - 1) / T), T, 0, stream>>>(
            h, src2, dst2, deg_src2, agg2, E2_);
    }

    // 7) agg2 *= norm_dst
    {
        int n = N2_ * HID_;
        gcn_scale_rows<<<(n + T - 1) / T, T, 0, stream>>>(agg2, deg_dst2, n, 8);
    }

    // 8) out = agg2 @ W2p + b2p   [M=10000, K=256, NB=48, Nc=47]
    {
        int tilesM = N2_ / 16, tilesN = OUTP_ / 16;
        int total = tilesM * tilesN;
        int blocks = (total + (T / 32) - 1) / (T / 32);
        gcn_gemm_wmma<HID_, OUTP_, false><<<blocks, T, 0, stream>>>(
            agg2, W2p, b2p, out, OUT_, tilesN, total);
    }
}